// O_Langevin_20340965114479
// MI455X (gfx1250) — compile-verified
//
#include <hip/hip_runtime.h>

// Constrained-Langevin step, dim=2, g(x)=|x|^2-1, H=2I (constant).
// Pure elementwise, HBM-bandwidth bound: stream 96MB via b128 loads/stores.

#define STEPSIZE   0.1f
#define SQRT_2H    0.44721359549995793f   // sqrt(2*STEPSIZE)
#define MCLIP      1000.0f

// Native clang vector types (required by __builtin_nontemporal_*; also
// guarantee b128/b64 vector memory ops).
typedef float v4f __attribute__((ext_vector_type(4)));
typedef float v2f __attribute__((ext_vector_type(2)));

__device__ __forceinline__ v2f step2(float x0, float x1, float e0, float e1) {
    // v = STEPSIZE * (-x) + sqrt(2*STEPSIZE) * xi
    float v0 = fmaf(-STEPSIZE, x0, SQRT_2H * e0);
    float v1 = fmaf(-STEPSIZE, x1, SQRT_2H * e1);

    // g, Dg, |Dg|^2
    float gx  = fmaf(x0, x0, fmaf(x1, x1, -1.0f));
    float Dg0 = 2.0f * x0;
    float Dg1 = 2.0f * x1;
    float dg2 = fmaf(Dg0, Dg0, Dg1 * Dg1);
    float inv = 1.0f / dg2;

    // project v off Dg
    float proj = fmaf(v0, Dg0, v1 * Dg1) * inv;
    float gp0  = fmaf(-proj, Dg0, v0);
    float gp1  = fmaf(-proj, Dg1, v1);

    // phi = ALPHA * sign(g) * |g|^BETA  with ALPHA=BETA=1  ->  phi = g (exact)
    float phi = gx;

    // Hessian correction, H = 2I (mirror reference structure):
    //   tDg = Dg/dg2 ; tH = tDg @ H = 2*tDg ; trH = 4
    //   term3 = 2*(tH . tDg) * Dg
    //   DxD   = tH + tDg*trH - term3
    float tDg0 = Dg0 * inv;
    float tDg1 = Dg1 * inv;
    float tH0  = 2.0f * tDg0;
    float tH1  = 2.0f * tDg1;
    float s3   = 2.0f * fmaf(tH0, tDg0, tH1 * tDg1);
    float DxD0 = fmaf(4.0f, tDg0, tH0) - s3 * Dg0;
    float DxD1 = fmaf(4.0f, tDg1, tH1) - s3 * Dg1;

    // dx = g_perp - STEPSIZE*phi*Dg/dg2 - STEPSIZE*DxD
    float c   = STEPSIZE * phi * inv;
    float dx0 = gp0 - c * Dg0 - STEPSIZE * DxD0;
    float dx1 = gp1 - c * Dg1 - STEPSIZE * DxD1;

    dx0 = fminf(fmaxf(dx0, -MCLIP), MCLIP);
    dx1 = fminf(fmaxf(dx1, -MCLIP), MCLIP);

    v2f o;
    o.x = x0 + dx0;
    o.y = x1 + dx1;
    return o;
}

// Main body: one v4f per thread = 2 particles. b128 loads, NT b128 stores.
__global__ void __launch_bounds__(256)
langevin_vec4(const v4f* __restrict__ x4, const v4f* __restrict__ xi4,
              v4f* __restrict__ out4, int n4) {
    int i = blockIdx.x * blockDim.x + threadIdx.x;
    if (i >= n4) return;

    v4f xv = x4[i];    // global_load_b128
    v4f ev = xi4[i];   // global_load_b128

    v2f a = step2(xv.x, xv.y, ev.x, ev.y);   // particle 2i
    v2f b = step2(xv.z, xv.w, ev.z, ev.w);   // particle 2i+1

    v4f o;
    o.x = a.x; o.y = a.y; o.z = b.x; o.w = b.y;

    // Output is write-once per replay; keep L2 for the resident inputs.
    __builtin_nontemporal_store(o, &out4[i]);  // global_store_b128 (NT)
}

// Tail: leftover particles (float pairs) past the float4-aligned region.
__global__ void langevin_tail(const v2f* __restrict__ x2,
                              const v2f* __restrict__ xi2,
                              v2f* __restrict__ out2,
                              int first_particle, int n_particles) {
    int p = first_particle + blockIdx.x * blockDim.x + threadIdx.x;
    if (p >= n_particles) return;
    v2f xv = x2[p];
    v2f ev = xi2[p];
    out2[p] = step2(xv.x, xv.y, ev.x, ev.y);
}

extern "C" void kernel_launch(void* const* d_in, const int* in_sizes, int n_in,
                              void* d_out, int out_size, void* d_ws, size_t ws_size,
                              hipStream_t stream) {
    const float* x   = (const float*)d_in[0];  // [N,2] float32
    const float* xi  = (const float*)d_in[1];  // [N,2] float32
    float*       out = (float*)d_out;          // [N,2] float32

    const long nfloat = in_sizes[0];           // N*2
    const int  n4     = (int)(nfloat / 4);     // v4f chunks (2 particles each)

    if (n4 > 0) {
        dim3 block(256);
        dim3 grid((n4 + 255) / 256);
        langevin_vec4<<<grid, block, 0, stream>>>(
            (const v4f*)x, (const v4f*)xi, (v4f*)out, n4);
    }

    // Possible tail of one particle (nfloat % 4 == 2); N=4M gives none, but
    // stay correct for any size.
    const int n_particles = (int)(nfloat / 2);
    const int done        = n4 * 2;
    if (done < n_particles) {
        int rem = n_particles - done;
        langevin_tail<<<(rem + 63) / 64, 64, 0, stream>>>(
            (const v2f*)x, (const v2f*)xi, (v2f*)out, done, n_particles);
    }
}